// ParticleGNN_29300266893915
// MI455X (gfx1250) — compile-verified
//
#include <hip/hip_runtime.h>
#include <math.h>

// Problem constants (from reference)
#define NNODES 16384   // B*N = 4*4096
#define NPTS   4096    // nodes per batch
#define DIMF   6
#define KNN    16
#define HEADS1 4
#define NEG_SLOPE 0.2f
#define BK     32      // K-chunk staged in LDS per step

typedef __attribute__((ext_vector_type(2))) float v2f;
typedef __attribute__((ext_vector_type(8))) float v8f;

#if __has_builtin(__builtin_amdgcn_global_load_async_to_lds_b32)
#define HAVE_ASYNC_LDS 1
typedef __attribute__((address_space(1))) int glob_i32;
typedef __attribute__((address_space(3))) int lds_i32;
#endif

__device__ __forceinline__ int imin(int a, int b) { return a < b ? a : b; }

__device__ __forceinline__ float gelu_exact(float v) {
  return 0.5f * v * (1.0f + erff(v * 0.7071067811865476f));
}

__device__ __forceinline__ void async_stage_wait() {
#ifdef HAVE_ASYNC_LDS
  asm volatile("s_wait_asynccnt 0" ::: "memory");
#endif
}

// ---------------------------------------------------------------------------
// 1) kNN: one block per destination node; distances in LDS; 16 argmin passes.
// ---------------------------------------------------------------------------
__global__ __launch_bounds__(256) void knn_kernel(const float* __restrict__ x,
                                                  int* __restrict__ idx_out) {
  __shared__ float dist[NPTS];
  __shared__ float rv[256];
  __shared__ int   ri[256];

  const int i  = blockIdx.x;
  const int b  = i >> 12;
  const int il = i & (NPTS - 1);
  const float* xb = x + (size_t)b * NPTS * DIMF;

  float xi[DIMF];
#pragma unroll
  for (int d = 0; d < DIMF; ++d) xi[d] = xb[(size_t)il * DIMF + d];

  for (int j = threadIdx.x; j < NPTS; j += blockDim.x) {
    float s = 0.0f;
#pragma unroll
    for (int d = 0; d < DIMF; ++d) {
      float t = xi[d] - xb[(size_t)j * DIMF + d];
      s += t * t;
    }
    dist[j] = (j == il) ? __builtin_inff() : s;
  }
  __syncthreads();

  for (int k = 0; k < KNN; ++k) {
    float bv = __builtin_inff();
    int   bi = -1;
    for (int j = threadIdx.x; j < NPTS; j += blockDim.x) {
      float v = dist[j];
      if (v < bv || (v == bv && (unsigned)j < (unsigned)bi)) { bv = v; bi = j; }
    }
    rv[threadIdx.x] = bv;
    ri[threadIdx.x] = bi;
    __syncthreads();
    for (int s = 128; s > 0; s >>= 1) {
      if (threadIdx.x < s) {
        float ov = rv[threadIdx.x + s];
        int   oi = ri[threadIdx.x + s];
        float cv = rv[threadIdx.x];
        int   ci = ri[threadIdx.x];
        if (ov < cv || (ov == cv && (unsigned)oi < (unsigned)ci)) {
          rv[threadIdx.x] = ov;
          ri[threadIdx.x] = oi;
        }
      }
      __syncthreads();
    }
    int w = ri[0];
    if (threadIdx.x == 0) {
      idx_out[(size_t)i * KNN + k] = b * NPTS + w;
      dist[w] = __builtin_inff();
    }
    __syncthreads();
  }
}

// ---------------------------------------------------------------------------
// 2) fp32 WMMA GEMM with LDS-staged (double-buffered) B tiles.
//    out = act(A[M,K] @ W[K,N] + bias + add)
//    LDS tile is PAIR-INTERLEAVED: tile[(k>>1)*128 + col*2 + (k&1)], so a
//    B fragment {W[k][col], W[k+1][col]} is one contiguous ds_load_b64.
//    Staging uses per-lane async b32 scatter (ASYNCcnt) when available.
// ---------------------------------------------------------------------------
__device__ __forceinline__ void stage_b(const float* __restrict__ W,
                                        float* __restrict__ tile, int kbase,
                                        int colbase, int Kd, int N, int tid) {
#ifdef HAVE_ASYNC_LDS
  if ((kbase + BK <= Kd) && (colbase + 64 <= N)) {
    // 2048 x b32 async transfers, 8 per thread, interleaved LDS scatter
    for (int e = tid; e < BK * 64; e += 256) {
      const int kk = e >> 6;
      const int c  = e & 63;
      const float* src = W + (size_t)(kbase + kk) * N + (colbase + c);
      float* dst = tile + ((kk >> 1) * 64 + c) * 2 + (kk & 1);
      __builtin_amdgcn_global_load_async_to_lds_b32(
          (glob_i32*)src, (lds_i32*)dst, 0, 0);
    }
    return;
  }
#endif
  // branchless fallback: clamped address + select-zero
  for (int e = tid; e < BK * 64; e += 256) {
    const int kk = e >> 6;
    const int c  = e & 63;
    const int gk = kbase + kk;
    const int gc = colbase + c;
    float v = W[(size_t)imin(gk, Kd - 1) * N + imin(gc, N - 1)];
    v = (gk < Kd && gc < N) ? v : 0.0f;
    tile[((kk >> 1) * 64 + c) * 2 + (kk & 1)] = v;
  }
}

__global__ __launch_bounds__(256) void wmma_gemm_kernel(
    const float* __restrict__ A, const float* __restrict__ W,
    const float* __restrict__ bias, const float* __restrict__ add,
    float* __restrict__ out, int M, int Kd, int N, int act) {
  __shared__ float btile[2][BK * 64];

  const int tid  = threadIdx.x;
  const int lane = tid & 31;
  const int wave = tid >> 5;
  const int half = lane >> 4;   // 0: lanes 0-15, 1: lanes 16-31
  const int r    = lane & 15;

  const int tileM   = blockIdx.x * 8 + wave;
  const int colbase = blockIdx.y * 64;
  const int row     = tileM * 16 + r;
  const int rowc    = imin(row, M - 1);           // clamp, no early return
  const float* aRow = A + (size_t)rowc * Kd;

  const int nchunks = (Kd + BK - 1) / BK;

  v8f acc[4];
#pragma unroll
  for (int t = 0; t < 4; ++t) acc[t] = (v8f){0, 0, 0, 0, 0, 0, 0, 0};

  stage_b(W, btile[0], 0, colbase, Kd, N, tid);
  async_stage_wait();
  __syncthreads();

  for (int c = 0; c < nchunks; ++c) {
    if (c + 1 < nchunks)
      stage_b(W, btile[(c + 1) & 1], (c + 1) * BK, colbase, Kd, N, tid);

    const float* tile = btile[c & 1];
    const int kbase   = c * BK;
    if (kbase + BK < Kd) __builtin_prefetch(aRow + kbase + BK, 0, 3);

    if (kbase + BK <= Kd) {
      // ---------- full chunk: no masks, pipelined A loads ----------
      const float* ap = aRow + kbase + 2 * half;
      v2f af[8];
#pragma unroll
      for (int s8 = 0; s8 < 8; ++s8)
        af[s8] = *(const v2f*)(ap + s8 * 4);     // clause of global_load_b64

#pragma unroll
      for (int s8 = 0; s8 < 8; ++s8) {
        const float* bp = tile + (s8 * 2 + half) * 128 + (r << 1);
        v2f b0 = *(const v2f*)(bp + 0);
        v2f b1 = *(const v2f*)(bp + 32);
        v2f b2 = *(const v2f*)(bp + 64);
        v2f b3 = *(const v2f*)(bp + 96);
        acc[0] = __builtin_amdgcn_wmma_f32_16x16x4_f32(false, af[s8], false, b0,
                                                       (short)0, acc[0], false, false);
        acc[1] = __builtin_amdgcn_wmma_f32_16x16x4_f32(false, af[s8], false, b1,
                                                       (short)0, acc[1], false, false);
        acc[2] = __builtin_amdgcn_wmma_f32_16x16x4_f32(false, af[s8], false, b2,
                                                       (short)0, acc[2], false, false);
        acc[3] = __builtin_amdgcn_wmma_f32_16x16x4_f32(false, af[s8], false, b3,
                                                       (short)0, acc[3], false, false);
      }
    } else {
      // ---------- tail chunk: clamped + select-zero A ----------
#pragma unroll
      for (int ks = 0; ks < BK; ks += 4) {
        const int k0g = kbase + ks + 2 * half;
        v2f av = *(const v2f*)(aRow + imin(k0g, Kd - 2));  // Kd even -> aligned
        av[0] = (k0g < Kd) ? av[0] : 0.0f;
        av[1] = (k0g + 1 < Kd) ? av[1] : 0.0f;

        const float* bp = tile + ((ks >> 1) + half) * 128 + (r << 1);
        v2f b0 = *(const v2f*)(bp + 0);
        v2f b1 = *(const v2f*)(bp + 32);
        v2f b2 = *(const v2f*)(bp + 64);
        v2f b3 = *(const v2f*)(bp + 96);
        acc[0] = __builtin_amdgcn_wmma_f32_16x16x4_f32(false, av, false, b0,
                                                       (short)0, acc[0], false, false);
        acc[1] = __builtin_amdgcn_wmma_f32_16x16x4_f32(false, av, false, b1,
                                                       (short)0, acc[1], false, false);
        acc[2] = __builtin_amdgcn_wmma_f32_16x16x4_f32(false, av, false, b2,
                                                       (short)0, acc[2], false, false);
        acc[3] = __builtin_amdgcn_wmma_f32_16x16x4_f32(false, av, false, b3,
                                                       (short)0, acc[3], false, false);
      }
    }

    async_stage_wait();
    __syncthreads();
  }

  // Epilogue: C/D layout -> col = base + (lane&15), row = v + 8*half
#pragma unroll
  for (int t = 0; t < 4; ++t) {
    const int col = colbase + t * 16 + r;
    if (col >= N) continue;
#pragma unroll
    for (int v = 0; v < 8; ++v) {
      const int orow = tileM * 16 + v + 8 * half;
      if (orow >= M) continue;
      float val = acc[t][v];
      if (bias) val += bias[col];
      if (add)  val += add[(size_t)orow * N + col];
      if (act)  val = gelu_exact(val);
      out[(size_t)orow * N + col] = val;
    }
  }
}

// ---------------------------------------------------------------------------
// 3) Attention logits: als[i,h] = <H[i,h,:], a_src[h,:]>, ald likewise.
// ---------------------------------------------------------------------------
__global__ __launch_bounds__(256) void attn_logits_kernel(
    const float* __restrict__ H, const float* __restrict__ a_src,
    const float* __restrict__ a_dst, float* __restrict__ als,
    float* __restrict__ ald, int heads, int C) {
  const int t = blockIdx.x * blockDim.x + threadIdx.x;
  if (t >= NNODES * heads) return;
  const int i = t / heads;
  const int h = t - i * heads;
  const float* hp = H + (size_t)i * heads * C + (size_t)h * C;
  const float* as = a_src + (size_t)h * C;
  const float* ad = a_dst + (size_t)h * C;
  float ss = 0.0f, sd = 0.0f;
  for (int c = 0; c < C; ++c) {
    float v = hp[c];
    ss += v * as[c];
    sd += v * ad[c];
  }
  als[t] = ss;
  ald[t] = sd;
}

// ---------------------------------------------------------------------------
// 4) GAT aggregation: softmax over 17 fixed sources, fused bias + GELU.
// ---------------------------------------------------------------------------
__global__ __launch_bounds__(256) void gat_aggregate_kernel(
    const float* __restrict__ H, const int* __restrict__ knn,
    const float* __restrict__ als, const float* __restrict__ ald,
    const float* __restrict__ bias, float* __restrict__ out,
    int heads, int C, int doGelu) {
  const int Ctot  = heads * C;
  const int npb   = 256 / Ctot;
  const int local = threadIdx.x / Ctot;
  const int t     = threadIdx.x - local * Ctot;
  const int i     = blockIdx.x * npb + local;
  if (i >= NNODES) return;
  const int h = t / C;

  int srcs[KNN + 1];
#pragma unroll
  for (int k = 0; k < KNN; ++k) srcs[k] = knn[(size_t)i * KNN + k];
  srcs[KNN] = i;

  const float ad = ald[(size_t)i * heads + h];

  float m = -__builtin_inff();
#pragma unroll
  for (int k = 0; k <= KNN; ++k) {
    float e = als[(size_t)srcs[k] * heads + h] + ad;
    e = (e > 0.0f) ? e : e * NEG_SLOPE;
    m = fmaxf(m, e);
  }

  float den = 0.0f, accv = 0.0f;
#pragma unroll
  for (int k = 0; k <= KNN; ++k) {
    const int s = srcs[k];
    float e = als[(size_t)s * heads + h] + ad;
    e = (e > 0.0f) ? e : e * NEG_SLOPE;
    const float w = expf(e - m);
    den  += w;
    accv += w * H[(size_t)s * Ctot + t];
  }

  float rr = accv / den + bias[t];
  if (doGelu) rr = gelu_exact(rr);
  out[(size_t)i * Ctot + t] = rr;
}

// ---------------------------------------------------------------------------
// Host launcher
// ---------------------------------------------------------------------------
static inline void run_gemm(const float* A, const float* W, const float* bias,
                            const float* add, float* out, int M, int K, int N,
                            int act, hipStream_t stream) {
  dim3 grid((M + 127) / 128, (N + 63) / 64);
  wmma_gemm_kernel<<<grid, 256, 0, stream>>>(A, W, bias, add, out, M, K, N, act);
}

extern "C" void kernel_launch(void* const* d_in, const int* in_sizes, int n_in,
                              void* d_out, int out_size, void* d_ws, size_t ws_size,
                              hipStream_t stream) {
  const float* x      = (const float*)d_in[0];
  const float* W1     = (const float*)d_in[1];
  const float* a_src1 = (const float*)d_in[2];
  const float* a_dst1 = (const float*)d_in[3];
  const float* b1     = (const float*)d_in[4];
  const float* W2     = (const float*)d_in[5];
  const float* a_src2 = (const float*)d_in[6];
  const float* a_dst2 = (const float*)d_in[7];
  const float* b2     = (const float*)d_in[8];
  const float* W3     = (const float*)d_in[9];
  const float* a_src3 = (const float*)d_in[10];
  const float* a_dst3 = (const float*)d_in[11];
  const float* b3     = (const float*)d_in[12];
  const float* res_W  = (const float*)d_in[13];
  const float* res_b  = (const float*)d_in[14];
  const float* m1_W   = (const float*)d_in[15];
  const float* m1_b   = (const float*)d_in[16];
  const float* m2_W   = (const float*)d_in[17];
  const float* m2_b   = (const float*)d_in[18];
  const float* m3_W   = (const float*)d_in[19];
  const float* m3_b   = (const float*)d_in[20];
  float* outF = (float*)d_out;

  // Workspace carve-up (~43 MB; stays L2-resident on MI455X's 192 MB L2)
  char* ws = (char*)d_ws;
  int*   idx  = (int*)ws;    ws += (size_t)NNODES * KNN * sizeof(int);
  float* als  = (float*)ws;  ws += (size_t)NNODES * HEADS1 * sizeof(float);
  float* ald  = (float*)ws;  ws += (size_t)NNODES * HEADS1 * sizeof(float);
  float* bufA = (float*)ws;  ws += (size_t)NNODES * 256 * sizeof(float);
  float* bufB = (float*)ws;  ws += (size_t)NNODES * 256 * sizeof(float);
  float* bufC = (float*)ws;  ws += (size_t)NNODES * 128 * sizeof(float);

  // 1) kNN graph
  knn_kernel<<<NNODES, 256, 0, stream>>>(x, idx);

  // 2) GAT layer 1
  run_gemm(x, W1, nullptr, nullptr, bufA, NNODES, 6, 256, 0, stream);
  attn_logits_kernel<<<(NNODES * 4 + 255) / 256, 256, 0, stream>>>(
      bufA, a_src1, a_dst1, als, ald, 4, 64);
  gat_aggregate_kernel<<<NNODES, 256, 0, stream>>>(bufA, idx, als, ald, b1,
                                                   bufB, 4, 64, 1);

  // 3) GAT layer 2
  run_gemm(bufB, W2, nullptr, nullptr, bufA, NNODES, 256, 256, 0, stream);
  attn_logits_kernel<<<(NNODES * 4 + 255) / 256, 256, 0, stream>>>(
      bufA, a_src2, a_dst2, als, ald, 4, 64);
  gat_aggregate_kernel<<<NNODES, 256, 0, stream>>>(bufA, idx, als, ald, b2,
                                                   bufB, 4, 64, 1);

  // 4) GAT layer 3 (1 head, 128 ch)
  run_gemm(bufB, W3, nullptr, nullptr, bufC, NNODES, 256, 128, 0, stream);
  attn_logits_kernel<<<(NNODES * 1 + 255) / 256, 256, 0, stream>>>(
      bufC, a_src3, a_dst3, als, ald, 1, 128);
  gat_aggregate_kernel<<<NNODES / 2, 256, 0, stream>>>(bufC, idx, als, ald, b3,
                                                       bufA, 1, 128, 1);

  // 5) y = gat3 + x @ res_W + res_b
  run_gemm(x, res_W, res_b, bufA, bufB, NNODES, 6, 128, 0, stream);

  // 6) MLP
  run_gemm(bufB, m1_W, m1_b, nullptr, bufC, NNODES, 128, 128, 1, stream);
  run_gemm(bufC, m2_W, m2_b, nullptr, bufA, NNODES, 128, 64, 1, stream);
  run_gemm(bufA, m3_W, m3_b, nullptr, outF, NNODES, 64, 6, 0, stream);
}